// Super_GLambaBlock_41154376630474
// MI455X (gfx1250) — compile-verified
//
#include <hip/hip_runtime.h>
#include <hip/hip_bf16.h>
#include <math.h>

// ---------------------------------------------------------------------------
// Bidirectional Mamba block for MI455X (gfx1250, wave32).
// All dense layers run through V_WMMA_F32_16X16X4_F32 (exact fp32 matmul),
// software-pipelined so next-step fragment loads overlap the matrix pipe.
// Scan is a per-(b,channel) register recurrence; the per-token B/C rows are
// broadcast into LDS with GLOBAL_LOAD_ASYNC_TO_LDS_B32 (ASYNCcnt-tracked).
// ---------------------------------------------------------------------------

#define DEV_INLINE __device__ __forceinline__

constexpr int B_ = 32, L_ = 196, D_ = 384;
constexpr int DI_ = 768, DS_ = 16, DTR_ = 24, KC_ = 4;
constexpr int LK_ = 16, LI_ = 384;
constexpr int NTOK = B_ * L_;            // 6272 tokens, multiple of 16
constexpr float EPS_ = 1e-5f;

typedef float v2f __attribute__((ext_vector_type(2)));
typedef float v8f __attribute__((ext_vector_type(8)));

#ifndef __has_builtin
#define __has_builtin(x) 0
#endif
#if __has_builtin(__builtin_amdgcn_wmma_f32_16x16x4_f32)
#define HAVE_WMMA_F32X4 1
#else
#define HAVE_WMMA_F32X4 0
#endif

DEV_INLINE float sigmoidf_(float x) { return 1.f / (1.f + __expf(-x)); }
DEV_INLINE float siluf_(float x)    { return x * sigmoidf_(x); }
DEV_INLINE float softplusf_(float x){ return (x > 20.f) ? x : log1pf(__expf(x)); }
DEV_INLINE float hswishf_(float x)  { float c = fminf(fmaxf(x + 3.f, 0.f), 6.f); return x * c * (1.f / 6.f); }

// Async global->LDS copy of one dword per lane (CDNA5 ASYNCcnt path).
// Must be executed by a fully-active wave; follow with async_wait0() in the
// same wave, then a block barrier before other waves consume the LDS data.
DEV_INLINE void async_copy_b32(const float* gsrc, float* lds_dst) {
#if defined(__gfx1250__)
  unsigned lds_off = (unsigned)(unsigned long long)(void*)lds_dst; // low 32 bits = LDS offset
  asm volatile("global_load_async_to_lds_b32 %0, %1, off"
               :: "v"(lds_off), "v"(gsrc) : "memory");
#else
  *lds_dst = *gsrc;  // host pass placeholder (never executed on device)
#endif
}
DEV_INLINE void async_wait0() {
#if defined(__gfx1250__)
  asm volatile("s_wait_asynccnt 0x0" ::: "memory");
#endif
}

// ---------------------------------------------------------------------------
// Generic GEMM: out[M,Ncols] = A[M,K] * W[Ncols,K]^T (+bias)(softplus?)(+resid)
// One wave per 16(M) x 64(N) tile, K-step 4 via V_WMMA_F32_16X16X4_F32.
// A fragment (16x4 f32): lanes 0-15 rows M=0..15 K={k,k+1}; lanes 16-31 K={k+2,k+3}.
// B fragment (4x16 f32): VGPR0 = rows K=k / k+2 across lane halves, VGPR1 = K+1.
// K-loop is software pipelined: iteration i's fragments are loaded during the
// WMMAs of iteration i-1 so the wave keeps >0 loads in flight at each wait.
// ---------------------------------------------------------------------------
__global__ __launch_bounds__(32) void gemm16x64_wmma(
    const float* __restrict__ A, int lda,
    const float* __restrict__ W,            // (Ncols, K) row-major
    const float* __restrict__ bias,         // nullable, per col
    const float* __restrict__ resid, int ldr,
    float* __restrict__ out, int ldo,
    int Ncols, int Kdim, int act)           // act: 0 none, 1 softplus
{
  const int lane    = threadIdx.x;          // 0..31
  const int rowBase = blockIdx.x * 16;
  const int colBase = blockIdx.y * 64;
  const int r16     = lane & 15;
  const int half    = lane >> 4;

  const int c0 = colBase + r16, c1 = c0 + 16, c2 = c0 + 32, c3 = c0 + 48;

#if HAVE_WMMA_F32X4
  // clamp OOB columns to a valid row of W (results masked at store)
  const float* w0 = W + (size_t)((c0 < Ncols) ? c0 : Ncols - 1) * Kdim;
  const float* w1 = W + (size_t)((c1 < Ncols) ? c1 : Ncols - 1) * Kdim;
  const float* w2 = W + (size_t)((c2 < Ncols) ? c2 : Ncols - 1) * Kdim;
  const float* w3 = W + (size_t)((c3 < Ncols) ? c3 : Ncols - 1) * Kdim;
  const float* arow = A + (size_t)(rowBase + r16) * lda;

  v8f acc0 = {}, acc1 = {}, acc2 = {}, acc3 = {};
  v2f a, b0, b1, b2, b3;
  {
    const int k0 = 2 * half;                 // prologue: k = 0
    a.x  = arow[k0]; a.y  = arow[k0 + 1];
    b0.x = w0[k0];   b0.y = w0[k0 + 1];
    b1.x = w1[k0];   b1.y = w1[k0 + 1];
    b2.x = w2[k0];   b2.y = w2[k0 + 1];
    b3.x = w3[k0];   b3.y = w3[k0 + 1];
  }
  for (int k = 4; k < Kdim; k += 4) {
    const int k0 = k + 2 * half;
    v2f an, b0n, b1n, b2n, b3n;              // issue next-step loads first
    an.x  = arow[k0]; an.y  = arow[k0 + 1];
    b0n.x = w0[k0];   b0n.y = w0[k0 + 1];
    b1n.x = w1[k0];   b1n.y = w1[k0 + 1];
    b2n.x = w2[k0];   b2n.y = w2[k0 + 1];
    b3n.x = w3[k0];   b3n.y = w3[k0 + 1];
    acc0 = __builtin_amdgcn_wmma_f32_16x16x4_f32(false, a, false, b0, (short)0, acc0, false, false);
    acc1 = __builtin_amdgcn_wmma_f32_16x16x4_f32(false, a, false, b1, (short)0, acc1, false, false);
    acc2 = __builtin_amdgcn_wmma_f32_16x16x4_f32(false, a, false, b2, (short)0, acc2, false, false);
    acc3 = __builtin_amdgcn_wmma_f32_16x16x4_f32(false, a, false, b3, (short)0, acc3, false, false);
    a = an; b0 = b0n; b1 = b1n; b2 = b2n; b3 = b3n;
  }
  acc0 = __builtin_amdgcn_wmma_f32_16x16x4_f32(false, a, false, b0, (short)0, acc0, false, false);
  acc1 = __builtin_amdgcn_wmma_f32_16x16x4_f32(false, a, false, b1, (short)0, acc1, false, false);
  acc2 = __builtin_amdgcn_wmma_f32_16x16x4_f32(false, a, false, b2, (short)0, acc2, false, false);
  acc3 = __builtin_amdgcn_wmma_f32_16x16x4_f32(false, a, false, b3, (short)0, acc3, false, false);

  // C/D layout: VGPR r -> (M=r, N=lane) for lanes 0-15, (M=r+8, N=lane-16) for 16-31
  #pragma unroll
  for (int r = 0; r < 8; ++r) {
    const int row = rowBase + (half ? r + 8 : r);
    float v[4] = { acc0[r], acc1[r], acc2[r], acc3[r] };
    const int cc[4] = { c0, c1, c2, c3 };
    #pragma unroll
    for (int t = 0; t < 4; ++t) {
      if (cc[t] < Ncols) {
        float vv = v[t];
        if (bias)      vv += bias[cc[t]];
        if (act == 1)  vv  = softplusf_(vv);
        if (resid)     vv += resid[(size_t)row * ldr + cc[t]];
        out[(size_t)row * ldo + cc[t]] = vv;
      }
    }
  }
#else
  // Scalar fallback (host pass / builtin-missing): same math, no WMMA.
  (void)half; (void)c0; (void)c1; (void)c2; (void)c3;
  for (int idx = lane; idx < 16 * 64; idx += 32) {
    const int row = rowBase + (idx >> 6);
    const int col = colBase + (idx & 63);
    if (col < Ncols) {
      const float* ar = A + (size_t)row * lda;
      const float* wr = W + (size_t)col * Kdim;
      float s = 0.f;
      for (int k = 0; k < Kdim; ++k) s += ar[k] * wr[k];
      if (bias)     s += bias[col];
      if (act == 1) s  = softplusf_(s);
      if (resid)    s += resid[(size_t)row * ldr + col];
      out[(size_t)row * ldo + col] = s;
    }
  }
#endif
}

// ---------------------------------------------------------------------------
__global__ __launch_bounds__(128) void layernorm_kernel(
    const float* __restrict__ x, const float* __restrict__ g,
    const float* __restrict__ b, float* __restrict__ out)
{
  __shared__ float s1[128], s2[128];
  const int n = blockIdx.x, tid = threadIdx.x;
  const float* row = x + (size_t)n * D_;
  float a = 0.f, q = 0.f;
  for (int i = tid; i < D_; i += 128) { float v = row[i]; a += v; q += v * v; }
  s1[tid] = a; s2[tid] = q; __syncthreads();
  for (int o = 64; o; o >>= 1) {
    if (tid < o) { s1[tid] += s1[tid + o]; s2[tid] += s2[tid + o]; }
    __syncthreads();
  }
  const float m  = s1[0] / D_;
  const float vr = s2[0] / D_ - m * m;
  const float r  = rsqrtf(vr + EPS_);
  for (int i = tid; i < D_; i += 128)
    out[(size_t)n * D_ + i] = (row[i] - m) * r * g[i] + b[i];
}

// causal depthwise conv (K=4) + SiLU, with optional sequence reversal (dir=1).
// Output is stored frame-ordered (position l in the direction's frame).
__global__ __launch_bounds__(256) void causal_conv_silu_kernel(
    const float* __restrict__ xz, const float* __restrict__ w,
    const float* __restrict__ bias, float* __restrict__ out, int dir)
{
  const int idx = blockIdx.x * 256 + threadIdx.x;
  if (idx >= NTOK * DI_) return;
  const int c = idx % DI_, n = idx / DI_;
  const int b = n / L_, l = n % L_;
  float acc = bias[c];
  #pragma unroll
  for (int k = 0; k < KC_; ++k) {
    const int j = l - (KC_ - 1) + k;
    if (j >= 0) {
      const int sj = dir ? (L_ - 1 - j) : j;
      acc += w[c * KC_ + k] * xz[(size_t)(b * L_ + sj) * (2 * DI_) + c];
    }
  }
  out[(size_t)n * DI_ + c] = siluf_(acc);
}

// Selective scan: one thread per (batch, channel), 16-wide state in registers.
// B/C rows (shared per token) broadcast through LDS via async-to-LDS copies
// issued by wave 0. Writes de-reversed output.
__global__ __launch_bounds__(256) void scan_kernel(
    const float* __restrict__ xz,     // [NTOK, 1536] original order (for z gate)
    const float* __restrict__ xconv,  // [NTOK, 768] frame order
    const float* __restrict__ xdbl,   // [NTOK, 64]  frame order (dt|B|C)
    const float* __restrict__ delta,  // [NTOK, 768] frame order
    const float* __restrict__ A_log,  // [768,16]
    const float* __restrict__ Dp,     // [768]
    float* __restrict__ y,            // [NTOK, 768] original order
    int dir, int accumulate)
{
  const int b = blockIdx.x / (DI_ / 256);
  const int c = (blockIdx.x % (DI_ / 256)) * 256 + threadIdx.x;
  float Ac[DS_];
  #pragma unroll
  for (int n = 0; n < DS_; ++n) Ac[n] = -__expf(A_log[c * DS_ + n]);
  const float Dc = Dp[c];
  float h[DS_];
  #pragma unroll
  for (int n = 0; n < DS_; ++n) h[n] = 0.f;

  __shared__ float sBC[2 * DS_];      // [0..15]=B row, [16..31]=C row
  for (int l = 0; l < L_; ++l) {
    const size_t row = (size_t)(b * L_ + l);
    if (threadIdx.x < 2 * DS_) {      // exactly wave 0, fully active
      async_copy_b32(&xdbl[row * 64 + DTR_ + threadIdx.x], &sBC[threadIdx.x]);
      async_wait0();
    }
    __syncthreads();

    if (l + 1 < L_) {                 // hint next step's operands (global_prefetch_b8)
      __builtin_prefetch(&delta[(row + 1) * DI_ + c], 0, 0);
      __builtin_prefetch(&xconv[(row + 1) * DI_ + c], 0, 0);
    }

    const float dl = delta[row * DI_ + c];
    const float xv = xconv[row * DI_ + c];
    const float dx = dl * xv;
    float yv = 0.f;
    #pragma unroll
    for (int n = 0; n < DS_; ++n) {
      h[n] = __expf(dl * Ac[n]) * h[n] + dx * sBC[n];
      yv  += h[n] * sBC[DS_ + n];
    }
    yv += xv * Dc;

    const int sl = dir ? (L_ - 1 - l) : l;
    const float z = xz[(size_t)(b * L_ + sl) * (2 * DI_) + DI_ + c];
    yv *= siluf_(z);
    const size_t orow = (size_t)(b * L_ + sl) * DI_ + c;
    if (accumulate) y[orow] += yv; else y[orow] = yv;
    __syncthreads();
  }
}

__global__ __launch_bounds__(256) void glu_kernel(
    const float* __restrict__ h, float* __restrict__ out)
{
  const int idx = blockIdx.x * 256 + threadIdx.x;
  if (idx >= NTOK * LI_) return;
  const int c = idx % LI_, n = idx / LI_;
  const float o  = h[(size_t)n * (2 * LI_) + c];
  const float gt = h[(size_t)n * (2 * LI_) + LI_ + c];
  out[(size_t)n * LI_ + c] = o * sigmoidf_(gt);
}

// depthwise "same" conv, LK=16, pad (7,8)
__global__ __launch_bounds__(256) void dwconv_same_kernel(
    const float* __restrict__ in, const float* __restrict__ w,
    const float* __restrict__ bias, float* __restrict__ out)
{
  const int idx = blockIdx.x * 256 + threadIdx.x;
  if (idx >= NTOK * LI_) return;
  const int c = idx % LI_, n = idx / LI_;
  const int b = n / L_, l = n % L_;
  float acc = bias[c];
  #pragma unroll
  for (int k = 0; k < LK_; ++k) {
    const int j = l - 7 + k;
    if (j >= 0 && j < L_) acc += w[c * LK_ + k] * in[(size_t)(b * L_ + j) * LI_ + c];
  }
  out[(size_t)n * LI_ + c] = acc;
}

// deterministic per-channel batch stats (mean over B*L), one block per channel
__global__ __launch_bounds__(256) void bnstats_kernel(
    const float* __restrict__ t, float* __restrict__ mean, float* __restrict__ rstd)
{
  const int c = blockIdx.x, tid = threadIdx.x;
  float s = 0.f, s2 = 0.f;
  for (int n = tid; n < NTOK; n += 256) {
    const float v = t[(size_t)n * LI_ + c];
    s += v; s2 += v * v;
  }
  __shared__ float ss[256], sq[256];
  ss[tid] = s; sq[tid] = s2; __syncthreads();
  for (int o = 128; o; o >>= 1) {
    if (tid < o) { ss[tid] += ss[tid + o]; sq[tid] += sq[tid + o]; }
    __syncthreads();
  }
  if (tid == 0) {
    const float m = ss[0] / NTOK;
    const float v = sq[0] / NTOK - m * m;
    mean[c] = m;
    rstd[c] = rsqrtf(v + EPS_);
  }
}

__global__ __launch_bounds__(256) void bnapply_hswish_kernel(
    const float* __restrict__ t, const float* __restrict__ mean,
    const float* __restrict__ rstd, const float* __restrict__ g,
    const float* __restrict__ b, float* __restrict__ out)
{
  const int idx = blockIdx.x * 256 + threadIdx.x;
  if (idx >= NTOK * LI_) return;
  const int c = idx % LI_;
  const float v = (t[idx] - mean[c]) * rstd[c] * g[c] + b[c];
  out[idx] = hswishf_(v);
}

// ---------------------------------------------------------------------------
extern "C" void kernel_launch(void* const* d_in, const int* in_sizes, int n_in,
                              void* d_out, int out_size, void* d_ws, size_t ws_size,
                              hipStream_t stream)
{
  (void)in_sizes; (void)n_in; (void)out_size; (void)ws_size;
  const float* x          = (const float*)d_in[0];
  const float* in_proj_w  = (const float*)d_in[1];
  const float* out_proj_w = (const float*)d_in[2];
  const float* A_log[2]   = {(const float*)d_in[3],  (const float*)d_in[10]};
  const float* conv_w[2]  = {(const float*)d_in[4],  (const float*)d_in[11]};
  const float* conv_b[2]  = {(const float*)d_in[5],  (const float*)d_in[12]};
  const float* xpw[2]     = {(const float*)d_in[6],  (const float*)d_in[13]};
  const float* dtw[2]     = {(const float*)d_in[7],  (const float*)d_in[14]};
  const float* dtb[2]     = {(const float*)d_in[8],  (const float*)d_in[15]};
  const float* Dp[2]      = {(const float*)d_in[9],  (const float*)d_in[16]};
  const float* lb_conv1_w = (const float*)d_in[17];
  const float* lb_conv1_b = (const float*)d_in[18];
  const float* lb_dw1_w   = (const float*)d_in[19];
  const float* lb_dw1_b   = (const float*)d_in[20];
  const float* lb_bn1_g   = (const float*)d_in[21];
  const float* lb_bn1_b   = (const float*)d_in[22];
  const float* lb_dw2_w   = (const float*)d_in[23];
  const float* lb_dw2_b   = (const float*)d_in[24];
  const float* lb_bn2_g   = (const float*)d_in[25];
  const float* lb_bn2_b   = (const float*)d_in[26];
  const float* lb_conv2_w = (const float*)d_in[27];
  const float* lb_conv2_b = (const float*)d_in[28];
  const float* ln1_g      = (const float*)d_in[29];
  const float* ln1_b      = (const float*)d_in[30];
  const float* ln2_g      = (const float*)d_in[31];
  const float* ln2_b      = (const float*)d_in[32];
  float* out = (float*)d_out;

  const size_t N = NTOK;
  float* ws        = (float*)d_ws;
  float* buf_h     = ws;                                 // N*384
  float* buf_xz    = buf_h     + N * (size_t)D_;         // N*1536
  float* buf_xconv = buf_xz    + N * (size_t)(2 * DI_);  // N*768
  float* buf_xdbl  = buf_xconv + N * (size_t)DI_;        // N*64
  float* buf_delta = buf_xdbl  + N * (size_t)64;         // N*768
  float* buf_y     = buf_delta + N * (size_t)DI_;        // N*768
  float* buf_x1    = buf_y     + N * (size_t)DI_;        // N*384
  float* buf_stats = buf_x1    + N * (size_t)D_;         // 4*384
  // reuse after the Mamba half finishes:
  float* buf_h2     = buf_h;
  float* buf_hsplit = buf_xz;                            // N*768 fits in N*1536
  float* buf_g1     = buf_xconv;
  float* buf_t1     = buf_delta;
  float* buf_u1     = buf_delta + N * (size_t)LI_;
  float* buf_t2     = buf_y;
  float* buf_u2     = buf_y + N * (size_t)LI_;
  float* mean1 = buf_stats, *rstd1 = buf_stats + LI_;
  float* mean2 = buf_stats + 2 * LI_, *rstd2 = buf_stats + 3 * LI_;

  const dim3 b32(32), b128(128), b256(256);
  const int rowTiles = NTOK / 16;                        // 392
  const int nElemBlk = (NTOK * LI_ + 255) / 256;
  const int nConvBlk = (NTOK * DI_ + 255) / 256;

  // LN1
  layernorm_kernel<<<NTOK, b128, 0, stream>>>(x, ln1_g, ln1_b, buf_h);
  // in_proj: [N,384] x [1536,384]^T -> xz [N,1536]
  gemm16x64_wmma<<<dim3(rowTiles, (2 * DI_) / 64), b32, 0, stream>>>(
      buf_h, D_, in_proj_w, nullptr, nullptr, 0, buf_xz, 2 * DI_, 2 * DI_, D_, 0);

  for (int dir = 0; dir < 2; ++dir) {
    causal_conv_silu_kernel<<<nConvBlk, b256, 0, stream>>>(
        buf_xz, conv_w[dir], conv_b[dir], buf_xconv, dir);
    // x_proj: [N,768] x [56,768]^T -> x_dbl [N,64-padded]
    gemm16x64_wmma<<<dim3(rowTiles, 1), b32, 0, stream>>>(
        buf_xconv, DI_, xpw[dir], nullptr, nullptr, 0, buf_xdbl, 64, DTR_ + 2 * DS_, DI_, 0);
    // dt_proj + softplus: [N,24] x [768,24]^T + b -> delta [N,768]
    gemm16x64_wmma<<<dim3(rowTiles, DI_ / 64), b32, 0, stream>>>(
        buf_xdbl, 64, dtw[dir], dtb[dir], nullptr, 0, buf_delta, DI_, DI_, DTR_, 1);
    scan_kernel<<<dim3(B_ * (DI_ / 256)), b256, 0, stream>>>(
        buf_xz, buf_xconv, buf_xdbl, buf_delta, A_log[dir], Dp[dir], buf_y, dir, dir);
  }

  // out_proj + residual: x1 = x + y @ W^T
  gemm16x64_wmma<<<dim3(rowTiles, D_ / 64), b32, 0, stream>>>(
      buf_y, DI_, out_proj_w, nullptr, x, D_, buf_x1, D_, D_, DI_, 0);

  // local block
  layernorm_kernel<<<NTOK, b128, 0, stream>>>(buf_x1, ln2_g, ln2_b, buf_h2);
  gemm16x64_wmma<<<dim3(rowTiles, (2 * LI_) / 64), b32, 0, stream>>>(
      buf_h2, D_, lb_conv1_w, lb_conv1_b, nullptr, 0, buf_hsplit, 2 * LI_, 2 * LI_, LI_, 0);
  glu_kernel<<<nElemBlk, b256, 0, stream>>>(buf_hsplit, buf_g1);
  dwconv_same_kernel<<<nElemBlk, b256, 0, stream>>>(buf_g1, lb_dw1_w, lb_dw1_b, buf_t1);
  bnstats_kernel<<<LI_, b256, 0, stream>>>(buf_t1, mean1, rstd1);
  bnapply_hswish_kernel<<<nElemBlk, b256, 0, stream>>>(buf_t1, mean1, rstd1, lb_bn1_g, lb_bn1_b, buf_u1);
  dwconv_same_kernel<<<nElemBlk, b256, 0, stream>>>(buf_u1, lb_dw2_w, lb_dw2_b, buf_t2);
  bnstats_kernel<<<LI_, b256, 0, stream>>>(buf_t2, mean2, rstd2);
  bnapply_hswish_kernel<<<nElemBlk, b256, 0, stream>>>(buf_t2, mean2, rstd2, lb_bn2_g, lb_bn2_b, buf_u2);
  // conv2 + residual -> final output
  gemm16x64_wmma<<<dim3(rowTiles, D_ / 64), b32, 0, stream>>>(
      buf_u2, LI_, lb_conv2_w, lb_conv2_b, buf_x1, D_, out, D_, D_, LI_, 0);
}